// GlowTTS_19825569038895
// MI455X (gfx1250) — compile-verified
//
#include <hip/hip_runtime.h>
#include <hip/hip_bf16.h>
#include <math.h>

#define NB   32
#define TX   512
#define TY   2048
#define DD   80
#define K2   160          // 2*DD, GEMM K dimension
#define KP   (K2/2)       // pair-packed K
#define NEGF (-1.0e9f)

typedef __attribute__((ext_vector_type(2))) float v2f;
typedef __attribute__((ext_vector_type(8))) float v8f;

// ---------------------------------------------------------------------------
// Token-side prep (WMMA *B* operand, pair-packed so a lane's (K,K+1) pair is
// one contiguous 8-byte load):
//   Apair[b][k/2][t][2] = [ exp(-2*x_logs) | x_m*exp(-2*x_logs) ] over k
// Also logp14[b,t] = sum_d( -0.5*log(2pi) - x_logs - 0.5*x_m^2*exp(-2*x_logs) )
// ---------------------------------------------------------------------------
__global__ void __launch_bounds__(128)
prep_tok_kernel(const float* __restrict__ x_m, const float* __restrict__ x_logs,
                float* __restrict__ Apair, float* __restrict__ logp14)
{
    __shared__ float red[128];
    const int bt = blockIdx.x;          // b*TX + t
    const int b  = bt / TX;
    const int t  = bt % TX;
    const int d  = threadIdx.x;
    float partial = 0.0f;
    if (d < DD) {
        float xl = x_logs[(size_t)bt * DD + d];
        float m  = x_m  [(size_t)bt * DD + d];
        float r  = __expf(-2.0f * xl);
        // k = d  -> value r ; k = DD + d -> value m*r
        Apair[(((size_t)b * KP + (d >> 1)) * TX + t) * 2 + (d & 1)]        = r;
        Apair[(((size_t)b * KP + ((DD + d) >> 1)) * TX + t) * 2 + (d & 1)] = m * r;
        partial = -0.91893853320467274178f - xl - 0.5f * m * m * r;
    }
    red[threadIdx.x] = partial;
    __syncthreads();
    for (int s = 64; s > 0; s >>= 1) {
        if (threadIdx.x < s) red[threadIdx.x] += red[threadIdx.x + s];
        __syncthreads();
    }
    if (threadIdx.x == 0) logp14[bt] = red[0];
}

// ---------------------------------------------------------------------------
// Frame-side prep (WMMA *A* operand, k contiguous per frame row):
//   Bt[b][y][k] = [ -0.5*z^2 | z ] over k
// ---------------------------------------------------------------------------
__global__ void __launch_bounds__(256)
prep_frm_kernel(const float* __restrict__ z, float* __restrict__ Bt)
{
    int i = blockIdx.x * blockDim.x + threadIdx.x;      // over B*DD*TY
    if (i >= NB * DD * TY) return;
    int y  = i % TY;
    int bd = i / TY;
    int d  = bd % DD;
    int b  = bd / DD;
    float v = z[i];                                     // coalesced in y
    Bt[((size_t)b * TY + y) * K2 + d]      = -0.5f * v * v;
    Bt[((size_t)b * TY + y) * K2 + DD + d] = v;
}

// ---------------------------------------------------------------------------
// logp GEMM with fp32 WMMA, computing f^T directly:
//   f[b][y][t] = sum_k Bt[b][y][k] * Atok[t][k]  (+ logp14[t], masked)
// M = Ty (frames), N = Tx (tokens).  Block = 4 waves; each wave owns a 16x64
// strip (one A fragment reused across 4 WMMAs per k-step).  All inner-loop
// loads are 8-byte (b64) and coalesced; epilogue stores are N-contiguous.
// ---------------------------------------------------------------------------
__global__ void __launch_bounds__(128)
logp_gemm_wmma(const float* __restrict__ Bt,      // [B][TY][K2]   (A operand)
               const float* __restrict__ Apair,   // [B][KP][TX][2] (B operand)
               const float* __restrict__ logp14,
               const float* __restrict__ x_mask, const float* __restrict__ z_mask,
               float* __restrict__ fbuf)
{
    const int lane  = threadIdx.x & 31;
    const int wave  = threadIdx.x >> 5;
    const int b     = blockIdx.z;
    const int tileM = blockIdx.y * 4 + wave;        // 16-row tile of TY
    const int nbase = blockIdx.x * 64;              // 64-col strip of TX

    const int my   = tileM * 16 + (lane & 15);      // frame row held by this lane
    const int koff = (lane >> 4) << 1;              // lanes 0-15: K,K+1 = k,k+1; 16-31: k+2,k+3
    const int nl   = nbase + (lane & 15);           // token col held by this lane

    v8f c0 = {}, c1 = {}, c2 = {}, c3 = {};
    const float* Arow = Bt + ((size_t)b * TY + my) * K2;
    const float* Bp   = Apair + (size_t)b * KP * TX * 2;

    for (int k = 0; k < K2; k += 4) {
        v2f a = *(const v2f*)(Arow + k + koff);
        const float* Bk = Bp + (size_t)((k + koff) >> 1) * TX * 2;
        v2f b0 = *(const v2f*)(Bk + 2 * (nl +  0));
        v2f b1 = *(const v2f*)(Bk + 2 * (nl + 16));
        v2f b2 = *(const v2f*)(Bk + 2 * (nl + 32));
        v2f b3 = *(const v2f*)(Bk + 2 * (nl + 48));
        c0 = __builtin_amdgcn_wmma_f32_16x16x4_f32(false, a, false, b0, (short)0, c0, false, false);
        c1 = __builtin_amdgcn_wmma_f32_16x16x4_f32(false, a, false, b1, (short)0, c1, false, false);
        c2 = __builtin_amdgcn_wmma_f32_16x16x4_f32(false, a, false, b2, (short)0, c2, false, false);
        c3 = __builtin_amdgcn_wmma_f32_16x16x4_f32(false, a, false, b3, (short)0, c3, false, false);
    }

    // D layout: VGPR r -> M(frame) = tileM*16 + r + 8*(lane>=16); N(token) = strip + sub*16 + (lane&15)
    const int mrow0 = tileM * 16 + ((lane >> 4) << 3);
    float* fb = fbuf + (size_t)b * TY * TX;
    #pragma unroll
    for (int sub = 0; sub < 4; ++sub) {
        v8f c = (sub == 0) ? c0 : (sub == 1) ? c1 : (sub == 2) ? c2 : c3;
        const int   N  = nbase + sub * 16 + (lane & 15);
        const float lp = logp14[b * TX + N];
        const float xm = x_mask[b * TX + N];
        #pragma unroll
        for (int r = 0; r < 8; ++r) {
            const int M = mrow0 + r;
            float val = c[r] + lp;
            if (xm * z_mask[b * TY + M] <= 0.0f) val = NEGF;
            fb[(size_t)M * TX + N] = val;          // N-contiguous across lanes
        }
    }
}

// ---------------------------------------------------------------------------
// Viterbi forward DP, in place over fbuf[b][y][x].
// f[x,y] = logp[x,y] + max( (x==0 ? (y==0?0:NEG) : f[x-1,y-1]),
//                           (x==y ? NEG : f[x,y-1]) )
// One block per batch, 512 threads, prev column in LDS.
// ---------------------------------------------------------------------------
__global__ void __launch_bounds__(512)
viterbi_fwd_kernel(float* __restrict__ fbuf)
{
    __shared__ float prev[TX];
    const int x = threadIdx.x;
    const int b = blockIdx.x;
    prev[x] = NEGF;
    __syncthreads();
    float* fb = fbuf + (size_t)b * TY * TX;
    for (int y = 0; y < TY; ++y) {
        float p_cur  = (x == y) ? NEGF : prev[x];
        float p_left = (x == 0) ? ((y == 0) ? 0.0f : NEGF) : prev[x - 1];
        __syncthreads();                       // reads done before overwrite
        float f = fb[(size_t)y * TX + x] + fmaxf(p_left, p_cur);
        fb[(size_t)y * TX + x] = f;
        prev[x] = f;
        __syncthreads();                       // writes visible for next y
    }
}

// ---------------------------------------------------------------------------
// Backtrack: one lane per batch walks y = Ty-1 .. 0 recording idxbuf[b][y]
// (-1 when inactive).
// ---------------------------------------------------------------------------
__global__ void __launch_bounds__(32)
viterbi_bwd_kernel(const float* __restrict__ fbuf,
                   const float* __restrict__ x_mask, const float* __restrict__ z_mask,
                   int* __restrict__ idxbuf)
{
    const int b = threadIdx.x;
    if (b >= NB) return;
    float s = 0.0f;
    for (int t = 0; t < TX; ++t) s += x_mask[b * TX + t];
    const int tx_len = (int)(s + 0.5f);
    s = 0.0f;
    for (int y = 0; y < TY; ++y) s += z_mask[b * TY + y];
    const int ty_len = (int)(s + 0.5f);

    int index = tx_len - 1;
    const float* fb = fbuf + (size_t)b * TY * TX;
    for (int y = TY - 1; y >= 0; --y) {
        const bool active = (y < ty_len);
        idxbuf[b * TY + y] = (active && index >= 0) ? index : -1;
        const int ic  = index > 0 ? index : 0;
        const int ym1 = y > 0 ? y - 1 : 0;
        const float f_here = fb[(size_t)ym1 * TX + ic];
        const float f_left = fb[(size_t)ym1 * TX + (ic > 0 ? ic - 1 : 0)];
        const bool move = (index != 0) && ((index == y) || (f_here < f_left));
        if (active && move) --index;
    }
}

// ---------------------------------------------------------------------------
// Scatter outputs: per (b, y) frame, gather token stats at x = idxbuf[b][y],
// write z_m / z_logs (zeros when inactive), set attn[b][x][y], accumulate
// durations.
// ---------------------------------------------------------------------------
__global__ void __launch_bounds__(128)
scatter_out_kernel(const int* __restrict__ idxbuf,
                   const float* __restrict__ x_m, const float* __restrict__ x_logs,
                   const float* __restrict__ x_mask, const float* __restrict__ z_mask,
                   float* __restrict__ z_m_out, float* __restrict__ z_logs_out,
                   float* __restrict__ attn_out, float* __restrict__ dur)
{
    const int by = blockIdx.x;            // b*TY + y
    const int b  = by / TY;
    const int y  = by % TY;
    const int x  = idxbuf[by];
    const int d  = threadIdx.x;
    float aval = 0.0f;
    if (x >= 0) aval = x_mask[b * TX + x] * z_mask[b * TY + y];
    if (d < DD) {
        float vm = 0.0f, vl = 0.0f;
        if (x >= 0) {
            vm = aval * x_m  [((size_t)b * TX + x) * DD + d];
            vl = aval * x_logs[((size_t)b * TX + x) * DD + d];
        }
        z_m_out   [((size_t)b * DD + d) * TY + y] = vm;
        z_logs_out[((size_t)b * DD + d) * TY + y] = vl;
    }
    if (threadIdx.x == 0 && x >= 0) {
        attn_out[((size_t)b * TX + x) * TY + y] = aval;
        atomicAdd(&dur[b * TX + x], aval);
    }
}

__global__ void __launch_bounds__(256)
logdur_kernel(const float* __restrict__ dur, const float* __restrict__ x_mask,
              float* __restrict__ out)
{
    int i = blockIdx.x * blockDim.x + threadIdx.x;
    if (i >= NB * TX) return;
    out[i] = logf(1e-8f + dur[i]) * x_mask[i];
}

// ---------------------------------------------------------------------------
extern "C" void kernel_launch(void* const* d_in, const int* in_sizes, int n_in,
                              void* d_out, int out_size, void* d_ws, size_t ws_size,
                              hipStream_t stream)
{
    const float* x_m    = (const float*)d_in[0];   // [B, TX, D]
    const float* x_logs = (const float*)d_in[1];   // [B, TX, D]
    const float* z      = (const float*)d_in[2];   // [B, D, TY]
    const float* x_mask = (const float*)d_in[3];   // [B, 1, TX]
    const float* z_mask = (const float*)d_in[4];   // [B, 1, TY]

    // Workspace carve-up
    float* Apair  = (float*)d_ws;                              // B*KP*TX*2 = B*K2*TX
    float* Bt     = Apair  + (size_t)NB * K2 * TX;             // B*TY*K2
    float* logp14 = Bt     + (size_t)NB * TY * K2;             // B*TX
    float* fbuf   = logp14 + (size_t)NB * TX;                  // B*TY*TX
    int*   idxbuf = (int*)(fbuf + (size_t)NB * TY * TX);       // B*TY
    float* dur    = (float*)(idxbuf + (size_t)NB * TY);        // B*TX

    // Output carve-up (return order: z_m, z_logs, log_durations_, attn)
    float* out        = (float*)d_out;
    float* z_m_out    = out;                                   // B*D*TY
    float* z_logs_out = z_m_out    + (size_t)NB * DD * TY;     // B*D*TY
    float* logdur_out = z_logs_out + (size_t)NB * DD * TY;     // B*TX
    float* attn_out   = logdur_out + (size_t)NB * TX;          // B*TX*TY

    // Zero the scatter targets (attn) and duration accumulators.
    hipMemsetAsync(attn_out, 0, (size_t)NB * TX * TY * sizeof(float), stream);
    hipMemsetAsync(dur, 0, (size_t)NB * TX * sizeof(float), stream);

    // 1) Build GEMM operands + per-token constant term.
    prep_tok_kernel<<<NB * TX, 128, 0, stream>>>(x_m, x_logs, Apair, logp14);
    prep_frm_kernel<<<(NB * DD * TY + 255) / 256, 256, 0, stream>>>(z, Bt);

    // 2) logp GEMM via fp32 WMMA -> fbuf[b][y][x] (written directly transposed).
    dim3 ggrid(TX / 64, TY / 64, NB);     // (8, 32, 32), 4 waves/block
    logp_gemm_wmma<<<ggrid, 128, 0, stream>>>(Bt, Apair, logp14, x_mask, z_mask, fbuf);

    // 3) Forward Viterbi DP (in place).
    viterbi_fwd_kernel<<<NB, 512, 0, stream>>>(fbuf);

    // 4) Backtrack.
    viterbi_bwd_kernel<<<1, 32, 0, stream>>>(fbuf, x_mask, z_mask, idxbuf);

    // 5) Outputs.
    scatter_out_kernel<<<NB * TY, 128, 0, stream>>>(idxbuf, x_m, x_logs, x_mask, z_mask,
                                                    z_m_out, z_logs_out, attn_out, dur);
    logdur_kernel<<<(NB * TX + 255) / 256, 256, 0, stream>>>(dur, x_mask, logdur_out);
}